// BaseDepthTransform_9294309228776
// MI455X (gfx1250) — compile-verified
//
#include <hip/hip_runtime.h>
#include <cstdint>

// ---- static problem config (mirrors the reference) ----
#define NCAM 6
#define ND   118
#define FHH  32
#define FWW  88
#define CCH  80
#define NPTS (NCAM*ND*FHH*FWW)      // 1,993,728 frustum points
#define NXV  360
#define NYV  360
#define SPATIAL (NXV*NYV)           // 129,600 BEV cells
#define TP   16                     // points per tile (NPTS % TP == 0)
#define CHUNKS (CCH/4)              // 20 float4 chunks per point
#define BS   (TP*CHUNKS)            // 320 threads per scatter block
#define NT   (NPTS/TP)              // 124,608 tiles

typedef float v2f __attribute__((ext_vector_type(2)));
typedef float v8f __attribute__((ext_vector_type(8)));

// -------------------- small 3x3 helpers --------------------
__device__ inline void inv3(const float* M, float* R) {
    float a=M[0],b=M[1],c=M[2],d=M[3],e=M[4],f=M[5],g=M[6],h=M[7],i=M[8];
    float A =  (e*i - f*h);
    float B = -(d*i - f*g);
    float Cc=  (d*h - e*g);
    float det = a*A + b*B + c*Cc;
    float s = 1.0f / det;
    R[0] =  A*s;            R[1] = -(b*i - c*h)*s;  R[2] =  (b*f - c*e)*s;
    R[3] =  B*s;            R[4] =  (a*i - c*g)*s;  R[5] = -(a*f - c*d)*s;
    R[6] =  Cc*s;           R[7] = -(a*h - b*g)*s;  R[8] =  (a*e - b*d)*s;
}
__device__ inline void mm3(const float* A, const float* B, float* C) {
    #pragma unroll
    for (int r = 0; r < 3; ++r)
        #pragma unroll
        for (int c = 0; c < 3; ++c)
            C[r*3+c] = A[r*3+0]*B[0*3+c] + A[r*3+1]*B[1*3+c] + A[r*3+2]*B[2*3+c];
}
__device__ inline void mv3(const float* A, const float* v, float* o) {
    #pragma unroll
    for (int r = 0; r < 3; ++r)
        o[r] = A[r*3+0]*v[0] + A[r*3+1]*v[1] + A[r*3+2]*v[2];
}

// -------------------- per-camera constant precompute --------------------
// cc layout per camera (24 floats):
//   [0..8]   M1 = inv(post_rots)            (stage-1 linear part)
//   [9..11]  b1 = -M1 @ post_trans          (stage-1 translation)
//   [12..20] A  = extra_rots@c2l_rots@inv(K)(stage-2 linear part)
//   [21..23] t  = extra_rots@c2l_trans + extra_trans
__global__ void k_pre(const float* __restrict__ c2lR, const float* __restrict__ c2lT,
                      const float* __restrict__ K,    const float* __restrict__ postR,
                      const float* __restrict__ postT,const float* __restrict__ eR,
                      const float* __restrict__ eT,   float* __restrict__ cc)
{
    int n = threadIdx.x;
    if (n >= NCAM) return;
    float iP[9], iK[9], b1[3], M2[9], A[9], t0[3];
    inv3(postR + n*9, iP);
    inv3(K + n*9, iK);
    mv3(iP, postT + n*3, b1);
    b1[0] = -b1[0]; b1[1] = -b1[1]; b1[2] = -b1[2];
    mm3(c2lR + n*9, iK, M2);
    mm3(eR, M2, A);
    mv3(eR, c2lT + n*3, t0);
    float* o = cc + n*24;
    #pragma unroll
    for (int i = 0; i < 9; ++i) o[i]      = iP[i];
    #pragma unroll
    for (int i = 0; i < 3; ++i) o[9+i]    = b1[i];
    #pragma unroll
    for (int i = 0; i < 9; ++i) o[12+i]   = A[i];
    #pragma unroll
    for (int i = 0; i < 3; ++i) o[21+i]   = t0[i] + eT[i];
}

// -------------------- WMMA geometry: per-point voxel index --------------------
// One wave transforms 16 points per iteration with two V_WMMA_F32_16X16X4_F32:
//   D1 = [M1|b1](rows 0..2 of A-operand) x B1(points as homogeneous columns)
//   perspective (u*d, v*d, d) per-lane on D1 columns
//   D2 = [A|t] x B2 -> lidar coords; truncate-voxelize exactly like the reference.
// A-operand (16x4 f32): V0 = {lanes0-15: K0, lanes16-31: K2}, V1 = {K1, K3}
// B-operand (4x16 f32): V0 = {lanes0-15: row0, lanes16-31: row2}, V1 = {row1, row3}
// D (16x16 f32):        VGPR r = {lanes0-15: M=r, N=lane; lanes16-31: M=r+8}
__global__ void __launch_bounds__(256) k_geom(const float* __restrict__ cc,
                                              int* __restrict__ idx, int nGroups)
{
    const int gtid = blockIdx.x * blockDim.x + threadIdx.x;
    const int lane = gtid & 31;
    const int m    = lane & 15;
    const int half = lane >> 4;
    const int wid  = gtid >> 5;                      // uniform within a wave
    const int nwav = (gridDim.x * blockDim.x) >> 5;

    for (int g = wid; g < nGroups; g += nwav) {
        const int p = g * 16 + m;                    // both halves mirror point m
        int tmp = p;
        const int wq = tmp % FWW; tmp /= FWW;
        const int hq = tmp % FHH; tmp /= FHH;
        const int dq = tmp % ND;  const int n = tmp / ND;  // n uniform per wave
        const float fx = (float)wq * (703.0f / 87.0f);     // linspace(0, IW-1, FW)
        const float fy = (float)hq * (255.0f / 31.0f);     // linspace(0, IH-1, FH)
        const float fd = 1.0f + 0.5f * (float)dq;          // arange(1, 60, 0.5)
        const float* c = cc + n*24;

        float a1x = 0.f, a1y = 0.f, a2x = 0.f, a2y = 0.f;
        if (m < 3) {                                   // only rows 0..2 are live
            a1x = half ? c[m*3+2]      : c[m*3+0];     // K2 : K0
            a1y = half ? c[9+m]        : c[m*3+1];     // b1 : K1
            a2x = half ? c[12+m*3+2]   : c[12+m*3+0];
            a2y = half ? c[21+m]       : c[12+m*3+1];
        }
        v2f A1 = {a1x, a1y};
        v2f A2 = {a2x, a2y};
        v2f B1 = {half ? fd : fx, half ? 1.0f : fy};
        v8f z  = {0.f,0.f,0.f,0.f,0.f,0.f,0.f,0.f};

        v8f D1 = __builtin_amdgcn_wmma_f32_16x16x4_f32(false, A1, false, B1,
                                                       (short)0, z, false, false);
        // lane n (<16) holds q = (D1[0], D1[1], D1[2]) for point n
        float qx = D1[0], qy = D1[1], qz = D1[2];
        float px = qx * qz, py = qy * qz, pz = qz;     // undo perspective
        px = __shfl(px, m, 32);                        // replicate low half -> both halves
        py = __shfl(py, m, 32);
        pz = __shfl(pz, m, 32);
        v2f B2 = {half ? pz : px, half ? 1.0f : py};

        v8f D2 = __builtin_amdgcn_wmma_f32_16x16x4_f32(false, A2, false, B2,
                                                       (short)0, z, false, false);
        if (half == 0) {
            const float gx = D2[0], gy = D2[1], gz = D2[2];
            // trunc-toward-zero casts, exactly like jnp astype(int32)
            const int cx = (int)((gx - (-53.85f - 0.15f)) / 0.3f);
            const int cy = (int)((gy - (-53.85f - 0.15f)) / 0.3f);
            const int cz = (int)((gz - (-10.0f)) / 20.0f);
            const bool kept = (cx >= 0) && (cx < NXV) && (cy >= 0) && (cy < NYV) && (cz == 0);
            idx[p] = kept ? (cx * NYV + cy) : -1;
        }
    }
}

// -------------------- zero fill --------------------
__global__ void k_zero(float4* __restrict__ p, int n4)
{
    const int stride = gridDim.x * blockDim.x;
    for (int i = blockIdx.x * blockDim.x + threadIdx.x; i < n4; i += stride)
        p[i] = make_float4(0.f, 0.f, 0.f, 0.f);
}

// -------------------- scatter-add with async-to-LDS double buffering ------------
// Each block handles TP=16 points x 20 float4 chunks. Features are streamed
// through LDS with global_load_async_to_lds_b128 (ASYNCcnt pipeline, 1 op per
// lane per tile -> s_wait_asynccnt 1 overlaps next tile's fetch with current
// tile's atomics). No barriers: each thread consumes only its own staged f4.
__global__ void __launch_bounds__(BS) k_scatter(const float* __restrict__ x,
                                                const int* __restrict__ idx,
                                                float* __restrict__ dst,
                                                int chanMajor, int nTiles)
{
    __shared__ __align__(16) float s_feat[2][BS * 4];
    const int tid   = threadIdx.x;
    const int pl    = tid / CHUNKS;
    const int chunk = tid - pl * CHUNKS;
    const unsigned lds0 = (unsigned)(uintptr_t)(&s_feat[0][tid * 4]);
    const unsigned lds1 = (unsigned)(uintptr_t)(&s_feat[1][tid * 4]);

    const int stride = gridDim.x;
    const int t0 = blockIdx.x;

    {   // prime the pipeline
        unsigned goff = ((unsigned)(t0 * TP + pl) * (unsigned)CCH + (unsigned)(chunk * 4)) * 4u;
        asm volatile("global_load_async_to_lds_b128 %0, %1, %2 offset:0"
                     :: "v"(lds0), "v"(goff), "s"(x) : "memory");
    }
    int par = 0;
    for (int t = t0; t < nTiles; t += stride) {
        const int tn = t + stride;
        if (tn < nTiles) {
            unsigned goff = ((unsigned)(tn * TP + pl) * (unsigned)CCH + (unsigned)(chunk * 4)) * 4u;
            unsigned lo = (par ^ 1) ? lds1 : lds0;
            asm volatile("global_load_async_to_lds_b128 %0, %1, %2 offset:0"
                         :: "v"(lo), "v"(goff), "s"(x) : "memory");
            asm volatile("s_wait_asynccnt 1" ::: "memory");   // current buffer ready
        } else {
            asm volatile("s_wait_asynccnt 0" ::: "memory");
        }
        const int s = idx[t * TP + pl];
        if (s >= 0) {
            const float4 v = *reinterpret_cast<const float4*>(&s_feat[par][tid * 4]);
            if (chanMajor) {  // direct into d_out [C][NX*NY]
                float* d = dst + (size_t)(chunk * 4) * SPATIAL + s;
                atomicAdd(d,               v.x);
                atomicAdd(d +     SPATIAL, v.y);
                atomicAdd(d + 2 * SPATIAL, v.z);
                atomicAdd(d + 3 * SPATIAL, v.w);
            } else {          // voxel-major accumulator [NX*NY][C] (L2-friendly)
                float* d = dst + (size_t)s * CCH + chunk * 4;
                atomicAdd(d + 0, v.x);
                atomicAdd(d + 1, v.y);
                atomicAdd(d + 2, v.z);
                atomicAdd(d + 3, v.w);
            }
        }
        par ^= 1;
    }
}

// -------------------- transpose [s][c] -> [c][s] --------------------
__global__ void k_final(const float* __restrict__ acc, float* __restrict__ out)
{
    const int i = blockIdx.x * blockDim.x + threadIdx.x;
    if (i >= CCH * SPATIAL) return;
    const int c = i / SPATIAL;          // consecutive i -> consecutive s: coalesced writes
    const int s = i - c * SPATIAL;      // reads stay hot in the 192 MB L2
    out[(size_t)c * SPATIAL + s] = acc[(size_t)s * CCH + c];
}

// -------------------- host --------------------
extern "C" void kernel_launch(void* const* d_in, const int* in_sizes, int n_in,
                              void* d_out, int out_size, void* d_ws, size_t ws_size,
                              hipStream_t stream)
{
    (void)in_sizes; (void)n_in; (void)out_size;
    const float* x     = (const float*)d_in[0];
    const float* c2lR  = (const float*)d_in[1];
    const float* c2lT  = (const float*)d_in[2];
    const float* intr  = (const float*)d_in[3];
    const float* postR = (const float*)d_in[4];
    const float* postT = (const float*)d_in[5];
    const float* eR    = (const float*)d_in[6];
    const float* eT    = (const float*)d_in[7];
    float* out = (float*)d_out;

    char*  ws  = (char*)d_ws;
    float* cc  = (float*)ws;                       // 144 floats of camera constants
    int*   idx = (int*)(ws + 1024);                // NPTS per-point voxel indices
    size_t accOff = (1024 + (size_t)NPTS * sizeof(int) + 255) & ~(size_t)255;
    float* accum  = (float*)(ws + accOff);
    const size_t accBytes = (size_t)CCH * SPATIAL * sizeof(float);  // 41.5 MB
    const bool useAccum = ws_size >= accOff + accBytes;

    k_pre<<<1, 32, 0, stream>>>(c2lR, c2lT, intr, postR, postT, eR, eT, cc);
    k_geom<<<512, 256, 0, stream>>>(cc, idx, NPTS / 16);

    const int sgrid = (NT < 2048) ? NT : 2048;
    if (useAccum) {
        k_zero<<<2048, 256, 0, stream>>>((float4*)accum, (CCH * SPATIAL) / 4);
        k_scatter<<<sgrid, BS, 0, stream>>>(x, idx, accum, 0, NT);
        k_final<<<(CCH * SPATIAL + 255) / 256, 256, 0, stream>>>(accum, out);
    } else {
        k_zero<<<2048, 256, 0, stream>>>((float4*)out, (CCH * SPATIAL) / 4);
        k_scatter<<<sgrid, BS, 0, stream>>>(x, idx, out, 1, NT);
    }
}